// ProgressivePatchExtractor_76579266888393
// MI455X (gfx1250) — compile-verified
//
#include <hip/hip_runtime.h>

// ProgressivePatchExtractor for MI455X (gfx1250).
// Pipeline:
//   for n in 0..3:  s1_partial_argmax<n> (32x32 blocks) -> s2_finalize (32 blocks)
//   extract_tdm (384 blocks x 1 wave): TDM tensor_load_to_lds / tensor_store_from_lds
//
// ws layout (bytes):
//   [0,    2048) : int   coords[32][4][4]  (x1,y1,x2,y2)
//   [2048, 6144) : float pvals [32][32]
//   [6144,10240) : int   pidx  [32][32]

#define HH      512
#define WW      512
#define EPS     128
#define HALF    64
#define MARGIN  32
#define NPATCH  4
#define NBATCH  32
#define NCH     3
#define SPLIT   32
#define CHUNK   ((HH * WW) / SPLIT)   // 8192 elements per block

typedef unsigned int u32;
typedef unsigned long long u64;
typedef __attribute__((ext_vector_type(4))) u32 u32x4;
typedef __attribute__((ext_vector_type(8))) u32 u32x8;

// ---------------------------------------------------------------------------
// Stage 1: per-block masked argmax over a contiguous chunk of one image's map.
// Templated on the greedy round so the box-mask test fully unrolls.
// ---------------------------------------------------------------------------
template <int N>
__global__ __launch_bounds__(256) void s1_partial_argmax(
    const float* __restrict__ umaps,      // [B,1,H,W]
    const int*  __restrict__ ws_coords,   // [B,4,4] (boxes selected so far)
    float* __restrict__ pvals,            // [B,SPLIT]
    int*   __restrict__ pidx)             // [B,SPLIT]
{
  __shared__ float sv[256];
  __shared__ int   si[256];
  __shared__ int   sbox[NPATCH * 4];

  const int tid   = threadIdx.x;
  const int chunk = blockIdx.x;
  const int b     = blockIdx.y;

  if (N > 0 && tid < 4 * N) sbox[tid] = ws_coords[b * 16 + tid];
  __syncthreads();

  const float* um = umaps + (size_t)b * (HH * WW) + (size_t)chunk * CHUNK;
  const int base_idx = chunk * CHUNK;

  float bv = -__builtin_inff();
  int   bi = 0x7fffffff;

  // 8 iterations of float4 per thread (8192 / 4 / 256).
  for (int j = 0; j < CHUNK / 4 / 256; ++j) {
    const int q = j * 256 + tid;                     // float4 index in chunk
    const float4 vv = ((const float4*)um)[q];
    const int idx0 = base_idx + q * 4;
    float vals[4] = {vv.x, vv.y, vv.z, vv.w};
#pragma unroll
    for (int e = 0; e < 4; ++e) {
      const int idx = idx0 + e;
      const int y = idx >> 9;          // / 512
      const int x = idx & (WW - 1);    // % 512
      bool masked = false;
#pragma unroll
      for (int k = 0; k < N; ++k) {
        const int x1 = sbox[k * 4 + 0], y1 = sbox[k * 4 + 1];
        const int x2 = sbox[k * 4 + 2], y2 = sbox[k * 4 + 3];
        masked |= (y >= y1 - MARGIN) && (y < y2 + MARGIN) &&
                  (x >= x1 - MARGIN) && (x < x2 + MARGIN);
      }
      const float v = masked ? -__builtin_inff() : vals[e];
      // indices visited in increasing order within a thread -> strict '>' keeps
      // the first (row-major) maximum, matching jnp.argmax.
      if (v > bv) { bv = v; bi = idx; }
    }
  }

  sv[tid] = bv; si[tid] = bi;
  __syncthreads();
  for (int off = 128; off > 0; off >>= 1) {
    if (tid < off) {
      const float v2 = sv[tid + off]; const int i2 = si[tid + off];
      if (v2 > sv[tid] || (v2 == sv[tid] && i2 < si[tid])) { sv[tid] = v2; si[tid] = i2; }
    }
    __syncthreads();
  }
  if (tid == 0) {
    pvals[b * SPLIT + chunk] = sv[0];
    pidx [b * SPLIT + chunk] = si[0];
  }
}

// ---------------------------------------------------------------------------
// Stage 2: combine the 32 partials of one image, emit box n.
// ---------------------------------------------------------------------------
__global__ __launch_bounds__(32) void s2_finalize(
    const float* __restrict__ pvals,
    const int*  __restrict__ pidx,
    int*   __restrict__ ws_coords,      // [B,4,4] int
    float* __restrict__ coords_out,     // [B,4,4] float (tail of d_out)
    int n)
{
  const int b = blockIdx.x;
  if (threadIdx.x != 0) return;

  float bv = -__builtin_inff();
  int   bi = 0x7fffffff;
  for (int k = 0; k < SPLIT; ++k) {
    const float v = pvals[b * SPLIT + k];
    const int   i = pidx [b * SPLIT + k];
    if (v > bv || (v == bv && i < bi)) { bv = v; bi = i; }
  }
  const int yc = bi >> 9;
  const int xc = bi & (WW - 1);
  int x1 = xc - HALF; if (x1 < 0) x1 = 0; if (x1 > WW - EPS) x1 = WW - EPS;
  int y1 = yc - HALF; if (y1 < 0) y1 = 0; if (y1 > HH - EPS) y1 = HH - EPS;

  const int base = b * 16 + n * 4;
  ws_coords[base + 0] = x1;        ws_coords[base + 1] = y1;
  ws_coords[base + 2] = x1 + EPS;  ws_coords[base + 3] = y1 + EPS;
  coords_out[base + 0] = (float)x1;         coords_out[base + 1] = (float)y1;
  coords_out[base + 2] = (float)(x1 + EPS); coords_out[base + 3] = (float)(y1 + EPS);
}

// ---------------------------------------------------------------------------
// Stage 3 (primary): patch extraction via the Tensor Data Mover.
// One block (one wave) per (b,n,c) plane. A single tensor_load_to_lds pulls
// the strided 128x128 f32 tile (row pitch 512 floats) into a packed 64 KB LDS
// buffer; after s_wait_tensorcnt 0, a single tensor_store_from_lds writes the
// packed tile to the output plane. D# bit layout per CDNA5 ISA ch.8.
// ---------------------------------------------------------------------------
__global__ __launch_bounds__(32) void extract_tdm(
    const float* __restrict__ images,     // [B,C,H,W]
    const int*  __restrict__ ws_coords,   // [B,4,4]
    float* __restrict__ patches_out)      // [B,N,C,EPS,EPS]
{
  __shared__ float tile[EPS * EPS];       // 64 KB staging tile

  const int plane = blockIdx.x;           // b*N*C + n*C + c
  const int c  = plane % NCH;
  const int bn = plane / NCH;             // b*N + n
  const int b  = bn / NPATCH;

  const int x1 = ws_coords[bn * 4 + 0];
  const int y1 = ws_coords[bn * 4 + 1];

  const u64 src = (u64)(size_t)(images + ((size_t)b * NCH + c) * (size_t)(HH * WW)
                                       + (size_t)y1 * WW + (size_t)x1);
  const u64 dst = (u64)(size_t)(patches_out + (size_t)plane * (EPS * EPS));
  // LDS byte address = low 32 bits of the generic address (LDS aperture rule).
  const u32 lds = (u32)(size_t)&tile[0];

  // ---- D# for the load (source tensor: 512x512 plane, tile 128x128) ----
  u32x4 g0; u32x8 g1;
  g0[0] = 1u;                                   // count=1, user descriptor
  g0[1] = lds;                                  // lds_addr [63:32]
  g0[2] = (u32)(src & 0xffffffffull);           // global_addr [95:64]
  g0[3] = (u32)((src >> 32) & 0x01ffffffull)    // global_addr [120:96]
          | 0x80000000u;                        // type=2 ("image") [127:126]
  g1[0] = 0x00020000u;                          // data_size=2 (4 bytes)
  g1[1] = (512u & 0xffffu) << 16;               // tensor_dim0 lo (atomic addr 0)
  g1[2] = (512u >> 16) | ((512u & 0xffffu) << 16); // td0 hi | tensor_dim1 lo
  g1[3] = (512u >> 16) | (128u << 16);          // td1 hi | tile_dim0=128
  g1[4] = 128u;                                 // tile_dim1=128 | tile_dim2=0
  g1[5] = 512u;                                 // tensor_dim0_stride lo32 = 512
  g1[6] = 0u;                                   // stride hi | dim1_stride lo
  g1[7] = 0u;                                   // dim1_stride hi (unused, 2D)
  asm volatile("tensor_load_to_lds %0, %1" :: "s"(g0), "s"(g1) : "memory");
  __builtin_amdgcn_s_wait_tensorcnt(0);         // tile resident in LDS

  // ---- D# for the store (dest tensor: packed 128x128 plane) ----
  u32x4 h0; u32x8 h1;
  h0[0] = 1u;
  h0[1] = lds;
  h0[2] = (u32)(dst & 0xffffffffull);
  h0[3] = (u32)((dst >> 32) & 0x01ffffffull) | 0x80000000u;
  h1[0] = 0x00020000u;                          // data_size=4B
  h1[1] = (128u & 0xffffu) << 16;               // tensor_dim0=128 lo
  h1[2] = (128u & 0xffffu) << 16;               // td0 hi=0 | tensor_dim1=128 lo
  h1[3] = (128u << 16);                         // td1 hi=0 | tile_dim0=128
  h1[4] = 128u;                                 // tile_dim1=128 | tile_dim2=0
  h1[5] = 128u;                                 // tensor_dim0_stride=128
  h1[6] = 0u;
  h1[7] = 0u;
  asm volatile("tensor_store_from_lds %0, %1" :: "s"(h0), "s"(h1) : "memory");
  __builtin_amdgcn_s_wait_tensorcnt(0);
}

// ---------------------------------------------------------------------------
// Stage 3 (fallback, compiled but not launched): async global<->LDS VMEM copy.
// Kept as the proven-to-assemble CDNA5 async path.
// ---------------------------------------------------------------------------
__global__ __launch_bounds__(256) void extract_async(
    const float* __restrict__ images,
    const int*  __restrict__ ws_coords,
    float* __restrict__ patches_out)
{
  __shared__ float smem[256 * 4];

  const int tid   = threadIdx.x;
  const int plane = blockIdx.x;
  const int c  = plane % NCH;
  const int bn = plane / NCH;
  const int b  = bn / NPATCH;

  const int x1 = ws_coords[bn * 4 + 0];
  const int y1 = ws_coords[bn * 4 + 1];

  const float* src = images + ((size_t)b * NCH + c) * (size_t)(HH * WW);
  float* dst = patches_out + (size_t)plane * (EPS * EPS);
  const u32 lds = (u32)(size_t)&smem[tid * 4];

  for (int it = 0; it < (EPS * EPS) / 4 / 256; ++it) {
    const int q   = it * 256 + tid;
    const int row = q >> 5;
    const int col = (q & 31) * 4;
    const u64 ga = (u64)(size_t)(src + (size_t)(y1 + row) * WW + (size_t)(x1 + col));
    const u64 da = (u64)(size_t)(dst + (size_t)q * 4);

    asm volatile("s_wait_asynccnt 0x0" ::: "memory");
    asm volatile("global_load_async_to_lds_b32 %0, %1, off" :: "v"(lds),       "v"(ga)       : "memory");
    asm volatile("global_load_async_to_lds_b32 %0, %1, off" :: "v"(lds + 4u),  "v"(ga + 4u)  : "memory");
    asm volatile("global_load_async_to_lds_b32 %0, %1, off" :: "v"(lds + 8u),  "v"(ga + 8u)  : "memory");
    asm volatile("global_load_async_to_lds_b32 %0, %1, off" :: "v"(lds + 12u), "v"(ga + 12u) : "memory");
    asm volatile("s_wait_asynccnt 0x0" ::: "memory");
    asm volatile("global_store_async_from_lds_b128 %0, %1, off" :: "v"(da), "v"(lds) : "memory");
  }
  asm volatile("s_wait_asynccnt 0x0" ::: "memory");
}

// ---------------------------------------------------------------------------
extern "C" void kernel_launch(void* const* d_in, const int* in_sizes, int n_in,
                              void* d_out, int out_size, void* d_ws, size_t ws_size,
                              hipStream_t stream) {
  (void)in_sizes; (void)n_in; (void)out_size; (void)ws_size;

  const float* images = (const float*)d_in[0];   // [32,3,512,512] f32
  const float* umaps  = (const float*)d_in[1];   // [32,1,512,512] f32

  float* out        = (float*)d_out;
  float* coords_out = out + (size_t)NBATCH * NPATCH * NCH * EPS * EPS;

  int*   ws_coords = (int*)d_ws;
  float* pvals     = (float*)((char*)d_ws + 2048);
  int*   pidx      = (int*)  ((char*)d_ws + 6144);

  const dim3 g1(SPLIT, NBATCH);
  s1_partial_argmax<0><<<g1, 256, 0, stream>>>(umaps, ws_coords, pvals, pidx);
  s2_finalize<<<NBATCH, 32, 0, stream>>>(pvals, pidx, ws_coords, coords_out, 0);
  s1_partial_argmax<1><<<g1, 256, 0, stream>>>(umaps, ws_coords, pvals, pidx);
  s2_finalize<<<NBATCH, 32, 0, stream>>>(pvals, pidx, ws_coords, coords_out, 1);
  s1_partial_argmax<2><<<g1, 256, 0, stream>>>(umaps, ws_coords, pvals, pidx);
  s2_finalize<<<NBATCH, 32, 0, stream>>>(pvals, pidx, ws_coords, coords_out, 2);
  s1_partial_argmax<3><<<g1, 256, 0, stream>>>(umaps, ws_coords, pvals, pidx);
  s2_finalize<<<NBATCH, 32, 0, stream>>>(pvals, pidx, ws_coords, coords_out, 3);

  extract_tdm<<<NBATCH * NPATCH * NCH, 32, 0, stream>>>(images, ws_coords, out);
}